// TopKRouter_70222715289755
// MI455X (gfx1250) — compile-verified
//
#include <hip/hip_runtime.h>
#include <hip/hip_bf16.h>

typedef __attribute__((ext_vector_type(2))) float v2f;
typedef __attribute__((ext_vector_type(4))) float v4f;
typedef __attribute__((ext_vector_type(8))) float v8f;

#define D_MODEL     4096
#define NUM_EXPERTS 64
#define LDS_STRIDE  66    // 64 + 2 pad -> bank-conflict-free per-lane row access
#define KCHUNK      16    // 4 WMMA K-steps per outer iteration
#define PF_DIST     512   // prefetch 512 floats (2KB) ahead per token row

// One wave32 per workgroup; wave computes logits for 32 tokens x 64 experts
// with V_WMMA_F32_16X16X4_F32, then per-lane softmax + top-2 over its token.
__global__ __launch_bounds__(32)
void topk_router_wmma(const float* __restrict__ x,
                      const float* __restrict__ W,
                      const float* __restrict__ bias,
                      float* __restrict__ probs_out,
                      float* __restrict__ topk_probs_out,
                      int*   __restrict__ topk_idx_out)
{
    __shared__ float lds[32 * LDS_STRIDE];

    const int lane = threadIdx.x & 31;
    const int half = lane >> 4;    // f32 WMMA A/B layout: low lanes K=0..1, high lanes K=2..3
    const int l16  = lane & 15;
    const int tok_base = blockIdx.x * 32;

    // Bias fragment per N-tile: this lane's expert column is 16*n + l16.
    float bias_frag[4];
#pragma unroll
    for (int n = 0; n < 4; ++n)
        bias_frag[n] = bias[16 * n + l16];

    v8f zero8 = {};
    v8f acc[2][4];
#pragma unroll
    for (int m = 0; m < 2; ++m)
#pragma unroll
        for (int n = 0; n < 4; ++n)
            acc[m][n] = zero8;

    // A-matrix 16x4 f32 layout: lane L<16 holds row M=L, {K, K+1}; lane>=16 holds {K+2, K+3}.
    const float* a0 = x + (size_t)(tok_base +      l16) * D_MODEL + 2 * half;
    const float* a1 = x + (size_t)(tok_base + 16 + l16) * D_MODEL + 2 * half;
    // B-matrix 4x16 f32, mirroring C layout: VGPR0 = rows {K, K+2}, VGPR1 = rows {K+1, K+3}.
    const float* bp = W + (size_t)(2 * half) * NUM_EXPERTS + l16;

    for (int k = 0; k < D_MODEL; k += KCHUNK) {
        // Software prefetch (global_prefetch_b8): pull the streams ahead into
        // cache without touching LOADcnt -> deep MLP for the HBM-bound stream.
        const int kp = k + PF_DIST;
        if (kp < D_MODEL) {
            __builtin_prefetch(a0 + kp, 0, 3);
            __builtin_prefetch(a1 + kp, 0, 3);
            __builtin_prefetch(bp + (size_t)kp * NUM_EXPERTS, 0, 3);
        }

#pragma unroll
        for (int kk = 0; kk < KCHUNK; kk += 4) {
            const int kc = k + kk;
            v2f af0 = *(const v2f*)(a0 + kc);           // global_load_b64, contiguous per lane
            v2f af1 = *(const v2f*)(a1 + kc);
            const float* bk = bp + (size_t)kc * NUM_EXPERTS;
            v2f bf[4];
#pragma unroll
            for (int n = 0; n < 4; ++n) {
                bf[n].x = bk[16 * n];
                bf[n].y = bk[16 * n + NUM_EXPERTS];
            }
#pragma unroll
            for (int n = 0; n < 4; ++n) {
                acc[0][n] = __builtin_amdgcn_wmma_f32_16x16x4_f32(
                    false, af0, false, bf[n], (short)0, acc[0][n], false, false);
                acc[1][n] = __builtin_amdgcn_wmma_f32_16x16x4_f32(
                    false, af1, false, bf[n], (short)0, acc[1][n], false, false);
            }
        }
    }

    // Spill logits (+bias) to LDS. C/D layout: VGPR r -> row r (lanes 0-15) / row r+8 (lanes 16-31).
#pragma unroll
    for (int m = 0; m < 2; ++m)
#pragma unroll
        for (int n = 0; n < 4; ++n)
#pragma unroll
            for (int r = 0; r < 8; ++r) {
                const int row = m * 16 + r + 8 * half;
                const int col = 16 * n + l16;
                lds[row * LDS_STRIDE + col] = acc[m][n][r] + bias_frag[n];
            }
    __syncthreads();   // single wave: just orders DS writes before reads

    // Per-lane: one token, 64 experts.
    float* myrow = &lds[lane * LDS_STRIDE];

    // Pass 1: running max == top-1, plus top-2 (strict '>' matches jax first-occurrence ties).
    float v1 = -__builtin_inff(), v2 = -__builtin_inff();
    int i1 = 0, i2 = 0;
#pragma unroll
    for (int e = 0; e < NUM_EXPERTS; ++e) {
        float v = myrow[e];
        if (v > v1)      { v2 = v1; i2 = i1; v1 = v; i1 = e; }
        else if (v > v2) { v2 = v;  i2 = e; }
    }
    const float mx = v1;

    // Pass 2: exp(v - max), in-place, accumulate sum.
    float sum = 0.f;
#pragma unroll
    for (int e = 0; e < NUM_EXPERTS; e += 2) {
        float e0 = __expf(myrow[e]     - mx);
        float e1 = __expf(myrow[e + 1] - mx);
        sum += e0 + e1;
        myrow[e]     = e0;
        myrow[e + 1] = e1;
    }
    const float inv = 1.f / sum;

    // Pass 3: stream normalized probs out (b128 stores).
    const size_t tok = (size_t)tok_base + lane;
    float* po = probs_out + tok * NUM_EXPERTS;
#pragma unroll
    for (int e = 0; e < NUM_EXPERTS; e += 4) {
        v4f p = { myrow[e] * inv, myrow[e + 1] * inv, myrow[e + 2] * inv, myrow[e + 3] * inv };
        *(v4f*)(po + e) = p;
    }

    // Top-2 renormalized (top-1 prob is exactly exp(0)/sum = inv).
    const float p1 = inv;
    const float p2 = __expf(v2 - mx) * inv;
    const float t  = p1 + p2 + 1e-9f;
    topk_probs_out[tok * 2 + 0] = p1 / t;
    topk_probs_out[tok * 2 + 1] = p2 / t;
    topk_idx_out[tok * 2 + 0]   = i1;
    topk_idx_out[tok * 2 + 1]   = i2;
}

extern "C" void kernel_launch(void* const* d_in, const int* in_sizes, int n_in,
                              void* d_out, int out_size, void* d_ws, size_t ws_size,
                              hipStream_t stream) {
    const float* x = (const float*)d_in[0];
    const float* W = (const float*)d_in[1];
    const float* b = (const float*)d_in[2];

    const int tokens = in_sizes[0] / D_MODEL;   // 16384 for the reference shapes

    float* probs  = (float*)d_out;
    float* topk_p = probs + (size_t)tokens * NUM_EXPERTS;
    int*   topk_i = (int*)(topk_p + (size_t)tokens * 2);

    dim3 grid(tokens / 32);
    dim3 block(32);
    hipLaunchKernelGGL(topk_router_wmma, grid, block, 0, stream,
                       x, W, b, probs, topk_p, topk_i);
}